// Memory_88055419502782
// MI455X (gfx1250) — compile-verified
//
#include <hip/hip_runtime.h>
#include <hip/hip_bf16.h>
#include <math.h>

// DNC memory step for MI455X (gfx1250, wave32).
// Memory-bound: 128MB link-matrix stream dominates (~6.5us at 23.3TB/s).
// Strategy: never materialize new_link; fuse link update + fwd/bwd einsums in
// one coalesced pass. f32 WMMA (16x16x4) for the interface GEMM. Per-batch
// LDS-resident kernels for softmax / bitonic-sort allocation / read mixing.

#define B_    32
#define INPUT 1024
#define N_    1024
#define WL_   64
#define R_    4
#define IFACE 471
#define IFACE_PAD 480
#define DELTA 1e-6f

typedef __attribute__((ext_vector_type(2))) float v2f;
typedef __attribute__((ext_vector_type(8))) float v8f;

// ---- workspace layout (floats) ----
#define OFF_XI    0                         // 32*480
#define OFF_RK    (OFF_XI   + B_*IFACE_PAD) // 32*256 normalized tanh read keys
#define OFF_RSTR  (OFF_RK   + B_*R_*WL_)    // 32*4
#define OFF_WK    (OFF_RSTR + B_*R_)        // 32*64 normalized tanh write key
#define OFF_WSTR  (OFF_WK   + B_*WL_)       // 32
#define OFF_ER    (OFF_WSTR + B_)           // 32*64 erase
#define OFF_WV    (OFF_ER   + B_*WL_)       // 32*64 write vector
#define OFF_FG    (OFF_WV   + B_*WL_)       // 32*4 free gates
#define OFF_AG    (OFF_FG   + B_*R_)        // 32
#define OFF_WG    (OFF_AG   + B_)           // 32
#define OFF_MODES (OFF_WG   + B_)           // 32*12
#define OFF_WW    (OFF_MODES+ B_*R_*3)      // 32*1024 new write weights
#define OFF_FWD   (OFF_WW   + B_*N_)        // 32*4*1024
#define OFF_BWD   (OFF_FWD  + B_*R_*N_)     // 32*4*1024
#define OFF_NM    (OFF_BWD  + B_*R_*N_)     // 32*1024*64 new memory

__device__ __forceinline__ float sigm(float x)  { return 1.0f / (1.0f + __expf(-x)); }
__device__ __forceinline__ float softp(float x) { return (x > 0.f) ? x + log1pf(__expf(-x)) : log1pf(__expf(x)); }

// ------------------------------------------------------------------
// K1: xi = x @ W_if + b_if via V_WMMA_F32_16X16X4_F32.
// Grid (30 n-tiles, 2 m-tiles), 32 threads = one wave per 16x16 tile.
// A frag 16x4: lanes 0-15 rows M, VGPR{0,1}=K{0,1}; lanes 16-31: K{2,3}.
// B frag 4x16: lanes 0-15 cols N, VGPR{0,1}=K{0,1}; lanes 16-31: K{2,3}.
// C/D 16x16: VGPR v, lanes 0-15: M=v, N=lane; lanes 16-31: M=v+8.
// Edge tile (cols 464..479 vs IFACE=471): loads use a clamped in-bounds
// address + value-level zero select (v_cndmask) -- no exec-mask branches
// in the hot loop.
// ------------------------------------------------------------------
__global__ void k_iface_gemm(const float* __restrict__ x,
                             const float* __restrict__ W,
                             const float* __restrict__ bias,
                             float* __restrict__ ws) {
    const int lane   = threadIdx.x;
    const int half   = lane >> 4;
    const int l15    = lane & 15;
    const int n_tile = blockIdx.x;     // 0..29
    const int m_tile = blockIdx.y;     // 0..1
    const int mrow   = m_tile * 16 + l15;
    const int col    = n_tile * 16 + l15;
    const bool colOK = (col < IFACE);
    const int colC   = colOK ? col : (IFACE - 1);   // clamped, always in-bounds
    const float msk  = colOK ? 1.f : 0.f;

    v8f acc = {0.f,0.f,0.f,0.f,0.f,0.f,0.f,0.f};
    for (int k = 0; k < INPUT; k += 4) {
        const int ka = k + half * 2;
        v2f a, bfrag;
        a.x = x[mrow * INPUT + ka];
        a.y = x[mrow * INPUT + ka + 1];
        bfrag.x = W[(size_t)ka * IFACE + colC] * msk;        // unconditional load
        bfrag.y = W[(size_t)(ka + 1) * IFACE + colC] * msk;  // + value select
        acc = __builtin_amdgcn_wmma_f32_16x16x4_f32(
            false, a, false, bfrag, (short)0, acc, false, false);
    }

    float* xi = ws + OFF_XI;
    if (colOK) {
        const float bv = bias[col];
        #pragma unroll
        for (int v = 0; v < 8; ++v) {
            const int row = m_tile * 16 + v + half * 8;
            xi[row * IFACE_PAD + col] = acc[v] + bv;
        }
    }
}

// ------------------------------------------------------------------
// K2: parse interface vector: activations, key normalization. 1 block/batch.
// ------------------------------------------------------------------
__global__ void k_parse(float* __restrict__ ws) {
    const int b = blockIdx.x, t = threadIdx.x;
    const float* xi = ws + OFF_XI + b * IFACE_PAD;
    __shared__ float sh_rk[256];
    __shared__ float sh_wk[64];
    __shared__ float sh_n[8];

    if (t < 256) sh_rk[t] = tanhf(xi[t]);               // read keys
    if (t < 64)  sh_wk[t] = tanhf(xi[260 + t]);         // write key
    __syncthreads();
    if (t < 4) {
        float ss = 0.f;
        for (int i = 0; i < 64; ++i) { float v = sh_rk[t*64+i]; ss += v*v; }
        sh_n[t] = sqrtf(ss) + DELTA;
    } else if (t == 4) {
        float ss = 0.f;
        for (int i = 0; i < 64; ++i) { float v = sh_wk[i]; ss += v*v; }
        sh_n[4] = sqrtf(ss) + DELTA;
    }
    __syncthreads();
    if (t < 256) ws[OFF_RK + b*256 + t] = sh_rk[t] / sh_n[t >> 6];
    if (t < 64) {
        ws[OFF_WK + b*64 + t] = sh_wk[t] / sh_n[4];
        ws[OFF_ER + b*64 + t] = sigm(xi[325 + t]);
        ws[OFF_WV + b*64 + t] = tanhf(xi[389 + t]);
    }
    if (t < 4) {
        ws[OFF_RSTR + b*4 + t] = softp(xi[256 + t]);
        ws[OFF_FG   + b*4 + t] = sigm(xi[453 + t]);
        // read_modes softmax over 3
        float m0 = xi[459 + t*3], m1 = xi[460 + t*3], m2 = xi[461 + t*3];
        float mx = fmaxf(m0, fmaxf(m1, m2));
        float e0 = __expf(m0-mx), e1 = __expf(m1-mx), e2 = __expf(m2-mx);
        float inv = 1.f / (e0 + e1 + e2);
        ws[OFF_MODES + b*12 + t*3 + 0] = e0 * inv;
        ws[OFF_MODES + b*12 + t*3 + 1] = e1 * inv;
        ws[OFF_MODES + b*12 + t*3 + 2] = e2 * inv;
    }
    if (t == 0) {
        ws[OFF_WSTR + b] = softp(xi[324]);
        ws[OFF_AG + b]   = sigm(xi[457]);
        ws[OFF_WG + b]   = sigm(xi[458]);
    }
}

// ------------------------------------------------------------------
// K3: per-batch (1024 threads): write content softmax, usage, bitonic-sort
// allocation, write weights, new_memory, zero fwd/bwd accumulators.
// ------------------------------------------------------------------
__global__ void k_write(const float* __restrict__ memory,
                        const float* __restrict__ read_w,
                        const float* __restrict__ write_w,
                        const float* __restrict__ usage,
                        float* __restrict__ ws) {
    const int b = blockIdx.x, t = threadIdx.x, n = t;
    __shared__ float s_f[1024];
    __shared__ float s_g[1024];
    __shared__ int   s_i[1024];
    __shared__ float s_wk[64], s_ev[64], s_wv[64];
    __shared__ float s_fg[4], s_sc[3];

    if (t < 64) {
        s_wk[t] = ws[OFF_WK + b*64 + t];
        s_ev[t] = ws[OFF_ER + b*64 + t];
        s_wv[t] = ws[OFF_WV + b*64 + t];
    }
    if (t < 4) s_fg[t] = ws[OFF_FG + b*4 + t];
    if (t == 0) { s_sc[0] = ws[OFF_WSTR + b]; s_sc[1] = ws[OFF_AG + b]; s_sc[2] = ws[OFF_WG + b]; }
    __syncthreads();

    // write content similarity
    float ss = 0.f, dot = 0.f;
    const float* mrow = memory + ((size_t)b * N_ + n) * WL_;
    #pragma unroll 8
    for (int w = 0; w < WL_; ++w) { float m = mrow[w]; ss += m*m; dot += m * s_wk[w]; }
    float sim = dot / (sqrtf(ss) + DELTA) * s_sc[0];

    // softmax over n
    s_g[t] = sim; __syncthreads();
    for (int s = 512; s > 0; s >>= 1) { if (t < s) s_g[t] = fmaxf(s_g[t], s_g[t+s]); __syncthreads(); }
    float mx = s_g[0]; __syncthreads();
    float e = __expf(sim - mx);
    s_g[t] = e; __syncthreads();
    for (int s = 512; s > 0; s >>= 1) { if (t < s) s_g[t] += s_g[t+s]; __syncthreads(); }
    float wcw = e / s_g[0]; __syncthreads();

    // usage
    float us = usage[b*N_ + n];
    float wwin = write_w[b*N_ + n];
    float uu = us + (1.f - us) * wwin;
    float psi = 1.f;
    #pragma unroll
    for (int r = 0; r < R_; ++r)
        psi *= (1.f - s_fg[r] * read_w[((size_t)b*R_ + r)*N_ + n]);
    float u = DELTA + (1.f - DELTA) * uu * psi;

    // bitonic sort ascending (key=u, payload=index)
    s_f[t] = u; s_i[t] = t;
    for (int k = 2; k <= 1024; k <<= 1) {
        for (int j = k >> 1; j > 0; j >>= 1) {
            __syncthreads();
            int ixj = t ^ j;
            if (ixj > t) {
                bool up = ((t & k) == 0);
                float a = s_f[t], c = s_f[ixj];
                bool sw = up ? (a > c) : (a < c);
                if (sw) {
                    s_f[t] = c; s_f[ixj] = a;
                    int tmp = s_i[t]; s_i[t] = s_i[ixj]; s_i[ixj] = tmp;
                }
            }
        }
    }
    __syncthreads();

    // exclusive prefix product of sorted u
    s_g[t] = s_f[t]; __syncthreads();
    for (int off = 1; off < 1024; off <<= 1) {
        float o = (t >= off) ? s_g[t - off] : 1.f;
        __syncthreads();
        s_g[t] *= o;
        __syncthreads();
    }
    float excl = (t > 0) ? s_g[t - 1] : 1.f;
    float alloc_sorted = (1.f - s_f[t]) * excl;
    __syncthreads();
    s_g[s_i[t]] = alloc_sorted;       // scatter = inverse permutation
    __syncthreads();
    float alloc = s_g[n];

    // write weights
    float ag = s_sc[1], wg = s_sc[2];
    float ww = wg * (ag * alloc + (1.f - ag) * wcw);
    ws[OFF_WW + b*N_ + n] = ww;

    // new memory
    float* nm = ws + OFF_NM + ((size_t)b * N_ + n) * WL_;
    #pragma unroll 8
    for (int w = 0; w < WL_; ++w)
        nm[w] = mrow[w] * (1.f - ww * s_ev[w]) + ww * s_wv[w];

    // zero fwd/bwd accumulators for this batch
    #pragma unroll
    for (int q = 0; q < 4; ++q) {
        ws[OFF_FWD + b*R_*N_ + t*4 + q] = 0.f;
        ws[OFF_BWD + b*R_*N_ + t*4 + q] = 0.f;
    }
}

// ------------------------------------------------------------------
// K4: fused link update + fwd/bwd einsums. Stream link_matrix exactly once.
// Grid (16 row-tiles, 32 batches) x 256 threads. Element (i,j):
//   nl = (1-ww[i]-ww[j])*L[i,j] + ww[i]*prec[j]   (0 on diagonal)
//   fwd[r,i] += nl*rw[r,j]   (wave-shuffle + LDS reduction per row)
//   bwd[r,j] += nl*rw[r,i]   (register accum over 64 rows, one atomic each)
// ------------------------------------------------------------------
__global__ void k_link(const float* __restrict__ L,
                       const float* __restrict__ prec,
                       const float* __restrict__ read_w,
                       float* __restrict__ ws) {
    const int t  = threadIdx.x;
    const int b  = blockIdx.y;
    const int i0 = blockIdx.x * 64;
    __shared__ float s_wwi[64];
    __shared__ float s_rwi[R_ * 64];
    __shared__ float s_red[8 * R_];

    const float* ww = ws + OFF_WW + b * N_;
    if (t < 64)  s_wwi[t] = ww[i0 + t];
    if (t < 256) {
        int r = t >> 6, ii = t & 63;
        s_rwi[t] = read_w[((size_t)b*R_ + r)*N_ + i0 + ii];
    }
    __syncthreads();

    float wwj[4], pj[4], rwj[4][R_], bacc[4][R_];
    #pragma unroll
    for (int c = 0; c < 4; ++c) {
        int j = c * 256 + t;
        wwj[c] = ww[j];
        pj[c]  = prec[b*N_ + j];
        #pragma unroll
        for (int r = 0; r < R_; ++r) {
            rwj[c][r]  = read_w[((size_t)b*R_ + r)*N_ + j];
            bacc[c][r] = 0.f;
        }
    }

    const int wid = t >> 5, lane = t & 31;
    for (int ii = 0; ii < 64; ++ii) {
        const int i = i0 + ii;
        const float wwi = s_wwi[ii];
        float rwi[R_];
        #pragma unroll
        for (int r = 0; r < R_; ++r) rwi[r] = s_rwi[r*64 + ii];

        const float* Lrow = L + ((size_t)b * N_ + i) * N_;
        float fp[R_] = {0.f, 0.f, 0.f, 0.f};
        #pragma unroll
        for (int c = 0; c < 4; ++c) {
            const int j = c * 256 + t;
            if (ii < 63) __builtin_prefetch(Lrow + N_ + j, 0, 0);  // next row -> global_prefetch_b8
            float l  = Lrow[j];
            float nl = (1.f - wwi - wwj[c]) * l + wwi * pj[c];
            nl = (i == j) ? 0.f : nl;
            #pragma unroll
            for (int r = 0; r < R_; ++r) {
                fp[r]      += nl * rwj[c][r];
                bacc[c][r] += nl * rwi[r];
            }
        }
        // reduce fp across the 256-thread block
        #pragma unroll
        for (int r = 0; r < R_; ++r)
            for (int off = 16; off > 0; off >>= 1)
                fp[r] += __shfl_xor(fp[r], off, 32);
        if (lane == 0)
            for (int r = 0; r < R_; ++r) s_red[wid*R_ + r] = fp[r];
        __syncthreads();
        if (t < R_) {
            float s = 0.f;
            #pragma unroll
            for (int w8 = 0; w8 < 8; ++w8) s += s_red[w8*R_ + t];
            ws[OFF_FWD + ((size_t)b*R_ + t)*N_ + i] = s;
        }
        __syncthreads();
    }

    #pragma unroll
    for (int c = 0; c < 4; ++c)
        #pragma unroll
        for (int r = 0; r < R_; ++r)
            atomicAdd(&ws[OFF_BWD + ((size_t)b*R_ + r)*N_ + c*256 + t], bacc[c][r]);
}

// ------------------------------------------------------------------
// K5: read content softmax on new_memory, mode mix, read_vectors -> d_out.
// ------------------------------------------------------------------
__global__ void k_read(float* __restrict__ ws, float* __restrict__ out) {
    const int b = blockIdx.x, t = threadIdx.x, n = t;
    __shared__ float s_rw[R_ * N_];      // 16KB new read weights
    __shared__ float s_red[1024];
    __shared__ float s_keys[R_ * WL_];
    __shared__ float s_str[R_], s_modes[R_ * 3];

    if (t < 256) s_keys[t] = ws[OFF_RK + b*256 + t];
    if (t < 4)   s_str[t]  = ws[OFF_RSTR + b*4 + t];
    if (t < 12)  s_modes[t]= ws[OFF_MODES + b*12 + t];
    __syncthreads();

    const float* nm = ws + OFF_NM + ((size_t)b * N_ + n) * WL_;
    float ss = 0.f, dot[R_] = {0.f, 0.f, 0.f, 0.f};
    #pragma unroll 8
    for (int w = 0; w < WL_; ++w) {
        float m = nm[w];
        ss += m * m;
        #pragma unroll
        for (int r = 0; r < R_; ++r) dot[r] += m * s_keys[r*WL_ + w];
    }
    float inv = 1.f / (sqrtf(ss) + DELTA);

    for (int r = 0; r < R_; ++r) {
        float sim = dot[r] * inv * s_str[r];
        s_red[t] = sim; __syncthreads();
        for (int s = 512; s > 0; s >>= 1) { if (t < s) s_red[t] = fmaxf(s_red[t], s_red[t+s]); __syncthreads(); }
        float mx = s_red[0]; __syncthreads();
        float e = __expf(sim - mx);
        s_red[t] = e; __syncthreads();
        for (int s = 512; s > 0; s >>= 1) { if (t < s) s_red[t] += s_red[t+s]; __syncthreads(); }
        float crw = e / s_red[0]; __syncthreads();

        float bw = ws[OFF_BWD + ((size_t)b*R_ + r)*N_ + n];
        float fw = ws[OFF_FWD + ((size_t)b*R_ + r)*N_ + n];
        s_rw[r*N_ + n] = s_modes[r*3+0]*bw + s_modes[r*3+1]*crw + s_modes[r*3+2]*fw;
    }
    __syncthreads();

    // read_vectors: (4 x 1024) @ (1024 x 64) per batch; 256 outputs.
    if (t < 256) {
        const int r = t >> 6, w = t & 63;
        const float* nmb = ws + OFF_NM + (size_t)b * N_ * WL_;
        float acc = 0.f;
        for (int n2 = 0; n2 < N_; ++n2)
            acc += s_rw[r*N_ + n2] * nmb[(size_t)n2 * WL_ + w];
        out[(b*R_ + r)*WL_ + w] = acc;
    }
}

extern "C" void kernel_launch(void* const* d_in, const int* in_sizes, int n_in,
                              void* d_out, int out_size, void* d_ws, size_t ws_size,
                              hipStream_t stream) {
    const float* x      = (const float*)d_in[0];
    const float* W_if   = (const float*)d_in[1];
    const float* b_if   = (const float*)d_in[2];
    const float* memory = (const float*)d_in[3];
    const float* link   = (const float*)d_in[4];
    const float* prec   = (const float*)d_in[5];
    const float* read_w = (const float*)d_in[6];
    const float* writew = (const float*)d_in[7];
    const float* usage  = (const float*)d_in[8];
    float* ws  = (float*)d_ws;
    float* out = (float*)d_out;

    k_iface_gemm<<<dim3(30, 2), 32, 0, stream>>>(x, W_if, b_if, ws);
    k_parse     <<<B_, 256, 0, stream>>>(ws);
    k_write     <<<B_, 1024, 0, stream>>>(memory, read_w, writew, usage, ws);
    k_link      <<<dim3(16, B_), 256, 0, stream>>>(link, prec, read_w, ws);
    k_read      <<<B_, 1024, 0, stream>>>(ws, out);
}